// STEIConvNetMScaleCompactGRAD_55662776156162
// MI455X (gfx1250) — compile-verified
//
#include <hip/hip_runtime.h>

#define N_NODES 50000
#define E_EDGES 800000
#define IN_DIM  16
#define H       32
#define EDGE_DIM 4
#define T_STEPS  4

typedef __attribute__((ext_vector_type(2))) float v2f;
typedef __attribute__((ext_vector_type(8))) float v8f;

__device__ __forceinline__ v8f wmma_f32_16x16x4(v2f a, v2f b, v8f c) {
  // 8 args: (neg_a, A, neg_b, B, c_mod, C, reuse_a, reuse_b)
  return __builtin_amdgcn_wmma_f32_16x16x4_f32(false, a, false, b, (short)0, c,
                                               false, false);
}

__device__ __forceinline__ void atomic_add_f32(float* p, float v) {
  __hip_atomic_fetch_add(p, v, __ATOMIC_RELAXED, __HIP_MEMORY_SCOPE_AGENT);
}

// ---------------------------------------------------------------------------
// h0 = inputs @ W_emb      [N,16] x [16,32] -> [N,32]
// One wave = 16 nodes x 32 cols (two 16x16 C tiles), K = 4 chunks of 4.
// ---------------------------------------------------------------------------
__global__ void embed_kernel(const float* __restrict__ inputs,
                             const float* __restrict__ W_emb,
                             float* __restrict__ h) {
  int wave = (int)((blockIdx.x * blockDim.x + threadIdx.x) >> 5);
  int lane = (int)(threadIdx.x & 31);
  if (wave >= N_NODES / 16) return;               // wave-uniform: EXEC stays full

  int row   = wave * 16 + (lane & 15);            // node index (A row)
  int khalf = (lane >> 4) * 2;                    // K sub-pair per lane half
  int colL  = lane & 15;                          // B/N column

  v8f c0 = {0.f,0.f,0.f,0.f,0.f,0.f,0.f,0.f};
  v8f c1 = c0;

#pragma unroll
  for (int kc = 0; kc < 4; ++kc) {
    int k0 = kc * 4 + khalf;
    v2f a;
    a.x = inputs[row * IN_DIM + k0];
    a.y = inputs[row * IN_DIM + k0 + 1];
    v2f b0, b1;
    b0.x = W_emb[k0 * H + colL];        b0.y = W_emb[(k0 + 1) * H + colL];
    b1.x = W_emb[k0 * H + 16 + colL];   b1.y = W_emb[(k0 + 1) * H + 16 + colL];
    c0 = wmma_f32_16x16x4(a, b0, c0);
    c1 = wmma_f32_16x16x4(a, b1, c1);
  }

  int mbase = wave * 16 + (lane >> 4) * 8;        // C layout: VGPR j -> M=j(+8)
  int n = lane & 15;
#pragma unroll
  for (int j = 0; j < 8; ++j) {
    h[(mbase + j) * H + n]      = c0[j];
    h[(mbase + j) * H + 16 + n] = c1[j];
  }
}

// ---------------------------------------------------------------------------
// Edge layer: m = relu([h[src]; ef_t] @ W_e)  then scatter-add into agg[dst].
// Persistent waves: B (9 K-chunks x 2 N-tiles) loaded ONCE into registers,
// then a grid-stride loop over 16-edge groups reuses them for every WMMA.
// ---------------------------------------------------------------------------
__global__ void edge_kernel(const float* __restrict__ h,
                            const float* __restrict__ e_feats, // [E, EDGE_DIM, T]
                            const float* __restrict__ W_e,     // [36, 32]
                            const int* __restrict__ src,
                            const int* __restrict__ dst,
                            float* __restrict__ agg,           // [N, 32] pre-zeroed
                            int t, int nwaves) {
  int wave  = (int)((blockIdx.x * blockDim.x + threadIdx.x) >> 5);
  int lane  = (int)(threadIdx.x & 31);
  int khalf = (lane >> 4) * 2;
  int colL  = lane & 15;

  // register-resident B tiles (loaded once per wave)
  v2f B0[9], B1[9];
#pragma unroll
  for (int kc = 0; kc < 9; ++kc) {
    int k0 = kc * 4 + khalf;
    B0[kc].x = W_e[k0 * H + colL];        B0[kc].y = W_e[(k0 + 1) * H + colL];
    B1[kc].x = W_e[k0 * H + 16 + colL];   B1[kc].y = W_e[(k0 + 1) * H + 16 + colL];
  }

  const int ngroups = E_EDGES / 16;                // 50000
  for (int g = wave; g < ngroups; g += nwaves) {   // wave-uniform loop
    int m = lane & 15;
    int e = g * 16 + m;
    int s = src[e];

    // prefetch next iteration's index lines (gfx1250 global_prefetch_b8)
    int gn = g + nwaves;
    if (gn < ngroups) {
      __builtin_prefetch(&src[gn * 16 + m], 0, 1);
      __builtin_prefetch(&dst[gn * 16 + m], 0, 1);
    }

    const float* hrow = h + (long)s * H;

    v8f c0 = {0.f,0.f,0.f,0.f,0.f,0.f,0.f,0.f};
    v8f c1 = c0;

#pragma unroll
    for (int kc = 0; kc < 8; ++kc) {               // K 0..31 : gathered h[src]
      int k0 = kc * 4 + khalf;
      v2f a;
      a.x = hrow[k0];
      a.y = hrow[k0 + 1];
      c0 = wmma_f32_16x16x4(a, B0[kc], c0);
      c1 = wmma_f32_16x16x4(a, B1[kc], c1);
    }
    {                                              // K 32..35 : edge features
      int k0 = 32 + khalf;
      v2f a;
      a.x = e_feats[((long)e * EDGE_DIM + (k0 - 32)) * T_STEPS + t];
      a.y = e_feats[((long)e * EDGE_DIM + (k0 - 31)) * T_STEPS + t];
      c0 = wmma_f32_16x16x4(a, B0[8], c0);
      c1 = wmma_f32_16x16x4(a, B1[8], c1);
    }

    int mrow = (lane >> 4) * 8;
    int n = lane & 15;
#pragma unroll
    for (int j = 0; j < 8; ++j) {
      int d = dst[g * 16 + mrow + j];
      float v0 = c0[j] > 0.f ? c0[j] : 0.f;        // relu fused into scatter
      float v1 = c1[j] > 0.f ? c1[j] : 0.f;
      atomic_add_f32(&agg[(long)d * H + n],      v0);
      atomic_add_f32(&agg[(long)d * H + 16 + n], v1);
    }
  }
}

// ---------------------------------------------------------------------------
// Node layer: h_out = relu([h; agg; rain_t] @ W_n)   K = 65 -> 17 chunks of 4
// chunk 16 covers K=64 (rain) with zero padding for K=65..67 (A and B).
// Persistent waves with register-resident B (17 x 2 x v2f).
// ---------------------------------------------------------------------------
__global__ void node_kernel(const float* __restrict__ h,
                            const float* __restrict__ agg,
                            const float* __restrict__ rain0,   // [N, T]
                            const float* __restrict__ W_n,     // [65, 32]
                            float* __restrict__ h_out,
                            int t, int nwaves) {
  int wave  = (int)((blockIdx.x * blockDim.x + threadIdx.x) >> 5);
  int lane  = (int)(threadIdx.x & 31);
  int khalf = (lane >> 4) * 2;
  int colL  = lane & 15;

  v2f B0[17], B1[17];
#pragma unroll
  for (int kc = 0; kc < 16; ++kc) {
    int k0 = kc * 4 + khalf;
    B0[kc].x = W_n[k0 * H + colL];        B0[kc].y = W_n[(k0 + 1) * H + colL];
    B1[kc].x = W_n[k0 * H + 16 + colL];   B1[kc].y = W_n[(k0 + 1) * H + 16 + colL];
  }
  // chunk 16: rows 64 valid, 65..67 are zero padding
  B0[16].x = (khalf == 0) ? W_n[64 * H + colL]      : 0.f;  B0[16].y = 0.f;
  B1[16].x = (khalf == 0) ? W_n[64 * H + 16 + colL] : 0.f;  B1[16].y = 0.f;

  const int ngroups = N_NODES / 16;                // 3125
  for (int g = wave; g < ngroups; g += nwaves) {   // wave-uniform loop
    int node = g * 16 + (lane & 15);

    v8f c0 = {0.f,0.f,0.f,0.f,0.f,0.f,0.f,0.f};
    v8f c1 = c0;

#pragma unroll
    for (int kc = 0; kc < 8; ++kc) {               // K 0..31 : h
      int k0 = kc * 4 + khalf;
      v2f a;
      a.x = h[(long)node * H + k0];
      a.y = h[(long)node * H + k0 + 1];
      c0 = wmma_f32_16x16x4(a, B0[kc], c0);
      c1 = wmma_f32_16x16x4(a, B1[kc], c1);
    }
#pragma unroll
    for (int kc = 8; kc < 16; ++kc) {              // K 32..63 : agg
      int k0 = kc * 4 + khalf;
      v2f a;
      a.x = agg[(long)node * H + (k0 - 32)];
      a.y = agg[(long)node * H + (k0 - 31)];
      c0 = wmma_f32_16x16x4(a, B0[kc], c0);
      c1 = wmma_f32_16x16x4(a, B1[kc], c1);
    }
    {                                              // K 64 : rain, 65..67 pad 0
      v2f a;
      a.x = (khalf == 0) ? rain0[(long)node * T_STEPS + t] : 0.f;
      a.y = 0.f;
      c0 = wmma_f32_16x16x4(a, B0[16], c0);
      c1 = wmma_f32_16x16x4(a, B1[16], c1);
    }

    int mbase = g * 16 + (lane >> 4) * 8;
    int n = lane & 15;
#pragma unroll
    for (int j = 0; j < 8; ++j) {
      float v0 = c0[j] > 0.f ? c0[j] : 0.f;
      float v1 = c1[j] > 0.f ? c1[j] : 0.f;
      h_out[(long)(mbase + j) * H + n]      = v0;
      h_out[(long)(mbase + j) * H + 16 + n] = v1;
    }
  }
}

// ---------------------------------------------------------------------------
// r[:, t] = h @ W_rain     (tiny: one thread per node)
// ---------------------------------------------------------------------------
__global__ void rain_kernel(const float* __restrict__ h,
                            const float* __restrict__ W_rain,  // [32,1]
                            float* __restrict__ out,           // [N, T]
                            int t) {
  int n = (int)(blockIdx.x * blockDim.x + threadIdx.x);
  if (n >= N_NODES) return;
  float acc = 0.f;
#pragma unroll
  for (int k = 0; k < H; ++k) acc += h[(long)n * H + k] * W_rain[k];
  out[(long)n * T_STEPS + t] = acc;
}

__global__ void zero_kernel(float* __restrict__ p, int count) {
  int i = (int)(blockIdx.x * blockDim.x + threadIdx.x);
  if (i < count) p[i] = 0.f;
}

// ---------------------------------------------------------------------------
extern "C" void kernel_launch(void* const* d_in, const int* in_sizes, int n_in,
                              void* d_out, int out_size, void* d_ws, size_t ws_size,
                              hipStream_t stream) {
  const float* inputs     = (const float*)d_in[0];
  const float* e_feats    = (const float*)d_in[1];
  const float* rain0      = (const float*)d_in[2];
  const float* W_emb      = (const float*)d_in[3];
  const float* W_edge_in  = (const float*)d_in[4];   // [2, 36, 32]
  const float* W_node_in  = (const float*)d_in[5];   // [2, 65, 32]
  const float* W_edge_out = (const float*)d_in[6];   // [36, 32]
  const float* W_node_out = (const float*)d_in[7];   // [65, 32]
  const float* W_rain     = (const float*)d_in[8];   // [32, 1]
  const int*   src        = (const int*)d_in[9];
  const int*   dst        = (const int*)d_in[10];
  float* out = (float*)d_out;

  // workspace: h ping-pong + agg, all L2-resident (3 * 6.4 MB)
  float* hA  = (float*)d_ws;
  float* hB  = hA + (long)N_NODES * H;
  float* agg = hB + (long)N_NODES * H;

  // embed: one wave per 16 nodes (one-shot, tiny)
  const int embed_blocks = ((N_NODES / 16) * 32 + 255) / 256;
  embed_kernel<<<embed_blocks, 256, 0, stream>>>(inputs, W_emb, hA);

  // persistent-wave grids
  const int edge_blocks = 512;                       // 4096 waves, ~12 groups each
  const int edge_waves  = edge_blocks * (256 / 32);
  const int node_blocks = 128;                       // 1024 waves, ~3 groups each
  const int node_waves  = node_blocks * (256 / 32);

  float* hcur = hA;
  float* hnxt = hB;
  for (int t = 0; t < T_STEPS; ++t) {
    for (int s = 0; s < 3; ++s) {
      const float* We = (s < 2) ? (W_edge_in + (long)s * 36 * H) : W_edge_out;
      const float* Wn = (s < 2) ? (W_node_in + (long)s * 65 * H) : W_node_out;
      zero_kernel<<<(N_NODES * H + 255) / 256, 256, 0, stream>>>(agg, N_NODES * H);
      edge_kernel<<<edge_blocks, 256, 0, stream>>>(hcur, e_feats, We, src, dst,
                                                   agg, t, edge_waves);
      node_kernel<<<node_blocks, 256, 0, stream>>>(hcur, agg, rain0, Wn, hnxt,
                                                   t, node_waves);
      float* tmp = hcur; hcur = hnxt; hnxt = tmp;
    }
    rain_kernel<<<(N_NODES + 255) / 256, 256, 0, stream>>>(hcur, W_rain, out, t);
  }
}